// MfdFC_21749714387073
// MI455X (gfx1250) — compile-verified
//
#include <hip/hip_runtime.h>

typedef __attribute__((ext_vector_type(2))) float v2f;
typedef __attribute__((ext_vector_type(8))) float v8f;

#define WFM_B    8
#define WFM_L    1024
#define WFM_CIN  32
#define WFM_COUT 32
#define WFM_D    8
#define WFM_NITER 3
#define WPB      8      // waves per block (wave32)
#define CROW     34     // padded LDS row stride (floats) for C: conflict-free b64 reads

// Wave-private LDS handoff: same-wave LDS ops are in-order (ISA ch11) and lanes
// are lockstep; this fence only constrains compiler reordering, no s_barrier.
#define WAVE_SYNC() __builtin_amdgcn_fence(__ATOMIC_ACQ_REL, "wavefront")

__global__ __launch_bounds__(WPB * 32)
void wfm_kernel(const float* __restrict__ x, const float* __restrict__ w,
                float* __restrict__ out) {
  __shared__ float s_wn[WFM_CIN * WFM_COUT];        // wn[i][o] (block-shared)
  __shared__ float s_inv[WFM_COUT];
  __shared__ float s_X[WPB][WFM_CIN][WFM_D];        // wave-private below here
  __shared__ float s_A[WPB][WFM_COUT][WFM_D];
  __shared__ float s_C[WPB][WFM_COUT * CROW];

  const int tid  = threadIdx.x;
  const int wv   = tid >> 5;
  const int lane = tid & 31;
  const int h    = lane >> 4;   // half-wave select (WMMA K/M split)
  const int ln   = lane & 15;

  const int p = blockIdx.x * WPB + wv;              // point index in [0, B*L)
  const float4* xp = (const float4*)(x + (size_t)p * (WFM_CIN * WFM_D));
  float4 xlo = xp[lane * 2 + 0];                    // x[p][lane][0:4]
  float4 xhi = xp[lane * 2 + 1];                    // x[p][lane][4:8]

  // ---- block-wide: wn = exp(w) normalized over input channels ----
  for (int e = tid; e < WFM_CIN * WFM_COUT; e += WPB * 32)
    s_wn[e] = __expf(w[e]);
  __syncthreads();
  if (tid < WFM_COUT) {
    float s = 0.f;
#pragma unroll
    for (int i = 0; i < WFM_CIN; ++i) s += s_wn[i * WFM_COUT + tid];
    s_inv[tid] = __builtin_amdgcn_rcpf(s);
  }
  __syncthreads();
  for (int e = tid; e < WFM_CIN * WFM_COUT; e += WPB * 32)
    s_wn[e] *= s_inv[e & (WFM_COUT - 1)];

  // stage X tile for this wave (wave-private from here on)
  *(float4*)&s_X[wv][lane][0] = xlo;
  *(float4*)&s_X[wv][lane][4] = xhi;
  __syncthreads();                                  // also covers s_wn writes

  // a_o := x_0 for all o
  {
    float4 lo = *(const float4*)&s_X[wv][0][0];
    float4 hi = *(const float4*)&s_X[wv][0][4];
    *(float4*)&s_A[wv][lane][0] = lo;
    *(float4*)&s_A[wv][lane][4] = hi;
  }
  WAVE_SYNC();

  // ---- loop-invariant WMMA operands (X-derived) ----
  // GEMM1 B operand: B[k][n] = x[16nt+n][4ks+k] ; V0=K(2h), V1=K(2h+1)
  v2f B1[2][2];
#pragma unroll
  for (int nt = 0; nt < 2; ++nt)
#pragma unroll
    for (int ks = 0; ks < 2; ++ks)
      B1[nt][ks] = *(const v2f*)&s_X[wv][16 * nt + ln][4 * ks + 2 * h];

  // GEMM2 A operand: A[m=d][k=i] = x[4ks+k][d]; rows d>=8 are zero padding
  v2f A2[8];
#pragma unroll
  for (int ks = 0; ks < 8; ++ks) {
    v2f a2; a2.x = 0.f; a2.y = 0.f;
    if (ln < WFM_D) {
      a2.x = s_X[wv][4 * ks + 2 * h + 0][ln];
      a2.y = s_X[wv][4 * ks + 2 * h + 1][ln];
    }
    A2[ks] = a2;
  }

  // wn gather matched to the G (C/D-matrix) lane layout: o=16mt+8h+r, i=16nt+ln
  float wnr[2][2][8];
#pragma unroll
  for (int mt = 0; mt < 2; ++mt)
#pragma unroll
    for (int nt = 0; nt < 2; ++nt) {
      const float* row = &s_wn[(16 * nt + ln) * WFM_COUT + 16 * mt + 8 * h];
#pragma unroll
      for (int r = 0; r < 8; ++r) wnr[mt][nt][r] = row[r];
    }

  const float CLIP = 1.0f - 1e-7f;

  for (int it = 0; it < WFM_NITER; ++it) {
    // GEMM1 A operand from current a: A[o][d], rows 16mt+ln
    v2f A1[2][2];
#pragma unroll
    for (int mt = 0; mt < 2; ++mt)
#pragma unroll
      for (int ks = 0; ks < 2; ++ks)
        A1[mt][ks] = *(const v2f*)&s_A[wv][16 * mt + ln][4 * ks + 2 * h];

    // G = A(32x8) x X^T(8x32) via 2x2 tiles, K=8 as 2 chained K=4 WMMAs
    v8f G[2][2];
#pragma unroll
    for (int mt = 0; mt < 2; ++mt)
#pragma unroll
      for (int nt = 0; nt < 2; ++nt) {
        v8f acc = {0.f, 0.f, 0.f, 0.f, 0.f, 0.f, 0.f, 0.f};
        acc = __builtin_amdgcn_wmma_f32_16x16x4_f32(false, A1[mt][0], false,
                                                    B1[nt][0], (short)0, acc,
                                                    false, false);
        acc = __builtin_amdgcn_wmma_f32_16x16x4_f32(false, A1[mt][1], false,
                                                    B1[nt][1], (short)0, acc,
                                                    false, false);
        G[mt][nt] = acc;
      }

    // elementwise: c[o,i] = wn[i,o] * theta/sin(theta)
    // After clipping, sin(theta) = sqrt(1-inner^2) > 1e-7 and theta > 1e-6,
    // so the reference's guards are dead: factor = acos(inner)*rsq(1-inner^2).
#pragma unroll
    for (int mt = 0; mt < 2; ++mt)
#pragma unroll
      for (int nt = 0; nt < 2; ++nt)
#pragma unroll
        for (int r = 0; r < 8; ++r) {
          float inner = G[mt][nt][r];
          inner = fminf(fmaxf(inner, -CLIP), CLIP);
          float theta = acosf(inner);
          float is = __builtin_amdgcn_rsqf(__builtin_fmaf(-inner, inner, 1.0f));
          float c = wnr[mt][nt][r] * theta * is;
          int o = 16 * mt + 8 * h + r;
          int i = 16 * nt + ln;
          s_C[wv][o * CROW + i] = c;
        }
    WAVE_SYNC();

    // grad0^T = X^T(8x32, M-padded to 16) x C^T(32x32): 2 N-tiles, 8 chained K=4
    v8f GT[2];
#pragma unroll
    for (int nt = 0; nt < 2; ++nt) {
      v8f acc = {0.f, 0.f, 0.f, 0.f, 0.f, 0.f, 0.f, 0.f};
#pragma unroll
      for (int ks = 0; ks < 8; ++ks) {
        // B[k][n] = C^T[4ks+k][16nt+n] = c[16nt+n][4ks+k]
        v2f b2 = *(const v2f*)&s_C[wv][(16 * nt + ln) * CROW + 4 * ks + 2 * h];
        acc = __builtin_amdgcn_wmma_f32_16x16x4_f32(false, A2[ks], false, b2,
                                                    (short)0, acc, false, false);
      }
      GT[nt] = acc;
    }
    WAVE_SYNC();  // s_C reads done; reuse as grad scratch below

    // Stage grad rows (lanes 0..15 hold grad0[o=16nt+ln][d]) so that every
    // lane runs exactly one spherical-exp chain for its own channel o=lane.
    float* gs = &s_C[wv][0];            // reused as [COUT][8] f32 scratch
    if (lane < 16) {
      float4 lo, hi;
      lo.x = GT[0][0]; lo.y = GT[0][1]; lo.z = GT[0][2]; lo.w = GT[0][3];
      hi.x = GT[0][4]; hi.y = GT[0][5]; hi.z = GT[0][6]; hi.w = GT[0][7];
      *(float4*)&gs[ln * 8 + 0] = lo;
      *(float4*)&gs[ln * 8 + 4] = hi;
      lo.x = GT[1][0]; lo.y = GT[1][1]; lo.z = GT[1][2]; lo.w = GT[1][3];
      hi.x = GT[1][4]; hi.y = GT[1][5]; hi.z = GT[1][6]; hi.w = GT[1][7];
      *(float4*)&gs[(ln + 16) * 8 + 0] = lo;
      *(float4*)&gs[(ln + 16) * 8 + 4] = hi;
    }
    WAVE_SYNC();

    // spherical exp for o = lane:
    // grad = P - <a,P> a  (tangent projection);  a' = cos|g| a + sinc|g| g
    {
      float P[8], a0[8];
      *(float4*)&P[0]  = *(const float4*)&gs[lane * 8 + 0];
      *(float4*)&P[4]  = *(const float4*)&gs[lane * 8 + 4];
      *(float4*)&a0[0] = *(const float4*)&s_A[wv][lane][0];
      *(float4*)&a0[4] = *(const float4*)&s_A[wv][lane][4];
      float s = 0.f;
#pragma unroll
      for (int d = 0; d < 8; ++d) s += a0[d] * P[d];
      float v[8];
      float n2 = 0.f;
#pragma unroll
      for (int d = 0; d < 8; ++d) {
        v[d] = P[d] - s * a0[d];
        n2 += v[d] * v[d];
      }
      bool small = n2 < 1e-12f;
      float n = __builtin_sqrtf(small ? 1.0f : n2);
      float cosn  = small ? 1.0f : __cosf(n);
      float sincn = small ? 1.0f : __sinf(n) * __builtin_amdgcn_rcpf(n);
      float4 lo, hi;
      lo.x = cosn * a0[0] + sincn * v[0];
      lo.y = cosn * a0[1] + sincn * v[1];
      lo.z = cosn * a0[2] + sincn * v[2];
      lo.w = cosn * a0[3] + sincn * v[3];
      hi.x = cosn * a0[4] + sincn * v[4];
      hi.y = cosn * a0[5] + sincn * v[5];
      hi.z = cosn * a0[6] + sincn * v[6];
      hi.w = cosn * a0[7] + sincn * v[7];
      *(float4*)&s_A[wv][lane][0] = lo;
      *(float4*)&s_A[wv][lane][4] = hi;
    }
    WAVE_SYNC();
  }

  // write result: row o = lane, contiguous 32B per channel
  float4* op = (float4*)(out + (size_t)p * (WFM_COUT * WFM_D));
  op[lane * 2 + 0] = *(const float4*)&s_A[wv][lane][0];
  op[lane * 2 + 1] = *(const float4*)&s_A[wv][lane][4];
}

extern "C" void kernel_launch(void* const* d_in, const int* in_sizes, int n_in,
                              void* d_out, int out_size, void* d_ws, size_t ws_size,
                              hipStream_t stream) {
  const float* x = (const float*)d_in[0];   // [B, L, CIN, D] f32
  const float* w = (const float*)d_in[1];   // [CIN, COUT]    f32
  float* out = (float*)d_out;               // [B, L, COUT, D] f32
  (void)in_sizes; (void)n_in; (void)out_size; (void)d_ws; (void)ws_size;

  dim3 grid((WFM_B * WFM_L) / WPB);         // 8192 points / 8 waves = 1024 blocks
  dim3 block(WPB * 32);                     // 8 wave32 waves
  hipLaunchKernelGGL(wfm_kernel, grid, block, 0, stream, x, w, out);
}